// EulerFlexRNNCell_40123584479480
// MI455X (gfx1250) — compile-verified
//
#include <hip/hip_runtime.h>

// CDNA5 / gfx1250: wave32, WMMA (not MFMA). We use V_WMMA_F32_16X16X4_F32
// since the reference recurrence is f32 and on-chip resident (compute bound,
// not HBM bound: ~23 GFLOP total vs ~170KB of weights).

typedef __attribute__((ext_vector_type(2))) float v2f;
typedef __attribute__((ext_vector_type(8))) float v8f;

#define DD 10
#define HH 64
#define BB 4096
#define NBATCH 256
#define DTs 0.015625f
#define NSTEPS 64

// ---- LDS layout (floats) ----
// fragW: packed B-fragments for z = [x(16 incl pad), h(64)] @ [[W],[U]]
//        [g][kc 0..19][nt 0..15][64]  (64 = 32 lanes x 2 floats)
#define FRAGW_ELEMS (2*20*16*64)       // 40960 floats = 160KB
// fragG: packed B-fragments for x-update GEMM, K = 88 (22 kc), one N-tile
#define FRAGG_ELEMS (2*22*64)          // 2816 floats = 11KB
// S: per-LSTM state matrix, 16 rows x [x(16 cols, 10 valid) | h(64 cols)] pad->84
#define S_STRIDE 84
#define SG (16*S_STRIDE)
#define S_ELEMS (2*SG)                 // 2688 floats
#define SMEM_FLOATS (FRAGW_ELEMS + FRAGG_ELEMS + S_ELEMS)   // 46464 -> 185856 B

__device__ __forceinline__ float sigf(float x) { return 1.0f / (1.0f + __expf(-x)); }

__device__ __forceinline__ v8f splat8(float x) {
  v8f r;
#pragma unroll
  for (int e = 0; e < 8; ++e) r[e] = x;
  return r;
}

__global__ __launch_bounds__(256) void euler_flex_rnn(
    const float* __restrict__ inputs,
    const float* __restrict__ rho0_r, const float* __restrict__ rho0_i,
    const float* __restrict__ Ar, const float* __restrict__ Ai,
    const float* __restrict__ Wr, const float* __restrict__ Ur, const float* __restrict__ br,
    const float* __restrict__ Wi, const float* __restrict__ Ui, const float* __restrict__ bi,
    const float* __restrict__ Dar_w, const float* __restrict__ Dar_b,
    const float* __restrict__ Dai_w, const float* __restrict__ Dai_b,
    float* __restrict__ out)
{
  extern __shared__ float smem[];
  float* fragW = smem;
  float* fragG = smem + FRAGW_ELEMS;
  float* S     = smem + FRAGW_ELEMS + FRAGG_ELEMS;

  const int tid  = threadIdx.x;
  const int lane = tid & 31;
  const int wave = tid >> 5;
  const int wq   = wave & 3;        // which 16-col slice of H this wave owns
  const int g    = wave >> 2;       // 0 = real LSTM, 1 = imag LSTM
  const int mrow = lane & 15;       // A-frag row, C-frag column for this lane
  const int sel  = (lane >> 4) << 1;// K sub-select (0 for lanes 0-15, 2 for 16-31)

  // ---------- one-time: pack weights into WMMA fragment layout ----------
  // B-fragment convention (16x16x4 f32): lane<16 holds {B[k0+0][n], B[k0+1][n]},
  // lane>=16 holds {B[k0+2][n], B[k0+3][n]}, n = lane&15.
  for (int idx = tid; idx < FRAGW_ELEMS; idx += 256) {
    int gg = idx / 20480; int r  = idx - gg * 20480;
    int kc = r >> 10;     int r2 = r & 1023;
    int nt = r2 >> 6;     int e  = r2 & 63;
    int ln = e >> 1;      int j  = e & 1;
    int n = nt * 16 + (ln & 15);
    int k = kc * 4 + ((ln >> 4) << 1) + j;   // K row: 0..9 = W, 16..79 = U
    float v = 0.0f;
    if (k < DD)       v = (gg ? Wi : Wr)[k * 256 + n];
    else if (k >= 16) v = (gg ? Ui : Ur)[(k - 16) * 256 + n];
    fragW[idx] = v;
  }
  // G_r = [Ar.T ; -Ai.T ; Dar_w], G_i = [Ai.T ; Ar.T ; Dai_w]; K rows:
  // 0..9 (px_r), 12..21 (px_i), 24..87 (h2); cols >= 10 zero.
  for (int idx = tid; idx < FRAGG_ELEMS; idx += 256) {
    int gg = idx / (22 * 64); int r = idx - gg * 22 * 64;
    int kc = r >> 6; int e = r & 63;
    int ln = e >> 1; int j = e & 1;
    int n = ln & 15;
    int k = kc * 4 + ((ln >> 4) << 1) + j;
    float v = 0.0f;
    if (n < DD) {
      if (k < DD)                      v = gg ? Ai[n * DD + k]        :  Ar[n * DD + k];
      else if (k >= 12 && k < 12 + DD) v = gg ? Ar[n * DD + (k - 12)] : -Ai[n * DD + (k - 12)];
      else if (k >= 24 && k < 24 + HH) v = (gg ? Dai_w : Dar_w)[(k - 24) * DD + n];
    }
    fragG[idx] = v;
  }
  // State init: x = rho0[triu], h = 0
  {
    const int triu[DD] = {0, 1, 2, 3, 5, 6, 7, 10, 11, 15};
    for (int idx = tid; idx < S_ELEMS; idx += 256) {
      int gg = idx / SG; int r = idx - gg * SG;
      int c = r % S_STRIDE;
      float v = 0.0f;
      if (c < DD) v = (gg ? rho0_i : rho0_r)[triu[c]];
      S[idx] = v;
    }
  }

  // per-lane constants
  const int   row = blockIdx.x * 16 + mrow;
  const float p0  = inputs[(row & (NBATCH - 1)) * 4] + 1e-8f;
  const float* bsrc = g ? bi : br;
  const float bz0 = bsrc[0 * 64 + wq * 16 + mrow];
  const float bz1 = bsrc[1 * 64 + wq * 16 + mrow];
  const float bz2 = bsrc[2 * 64 + wq * 16 + mrow];
  const float bz3 = bsrc[3 * 64 + wq * 16 + mrow];
  const float bx  = (mrow < DD) ? (g ? Dai_b : Dar_b)[mrow] : 0.0f;

  v8f creg = splat8(0.0f);            // cell state, resident in registers
  float* Sg = S + g * SG;
  const float* Srow = Sg + mrow * S_STRIDE;  // this lane's A-fragment row
  const float* wb = fragW + (g * 320 + wq) * 64 + (lane << 1);

  __syncthreads();

#pragma unroll 1
  for (int step = 0; step < NSTEPS; ++step) {
    // ---- z = [x,h] @ [[W],[U]] + b : 16x256 via 4 owned N-tiles (i,f,g,o) ----
    v8f zi = splat8(bz0), zf = splat8(bz1), zg = splat8(bz2), zo = splat8(bz3);
#pragma unroll
    for (int kc = 0; kc < 20; ++kc) {
      v2f a = *(const v2f*)(Srow + kc * 4 + sel);   // A frag (shared by 4 tiles)
      const float* wk = wb + kc * 1024;
      zi = __builtin_amdgcn_wmma_f32_16x16x4_f32(false, a, false, *(const v2f*)(wk),          (short)0, zi, false, false);
      zf = __builtin_amdgcn_wmma_f32_16x16x4_f32(false, a, false, *(const v2f*)(wk + 256),    (short)0, zf, false, false);
      zg = __builtin_amdgcn_wmma_f32_16x16x4_f32(false, a, false, *(const v2f*)(wk + 512),    (short)0, zg, false, false);
      zo = __builtin_amdgcn_wmma_f32_16x16x4_f32(false, a, false, *(const v2f*)(wk + 768),    (short)0, zo, false, false);
    }

    // ---- LSTM gates: fully in registers (same (v,lane)->(m,n) frag layout) ----
    v8f h2;
#pragma unroll
    for (int e = 0; e < 8; ++e) {
      float c2 = sigf(zf[e]) * creg[e] + sigf(zi[e]) * tanhf(zg[e]);
      creg[e] = c2;
      h2[e]   = sigf(zo[e]) * tanhf(c2);
    }

    __syncthreads();  // (A) all waves done reading old S

    // write h2 into S h-region, cols [16+16*wq, 16+16*wq+16)
    {
      float* hdst = Sg + 16 + wq * 16 + (lane & 15);
#pragma unroll
      for (int v = 0; v < 8; ++v) {
        int mm = v + ((lane >> 4) << 3);
        hdst[mm * S_STRIDE] = h2[v];
      }
    }
    __syncthreads();  // (B) h2 visible

    // ---- x-update GEMM: [p0*xr, p0*xi, h2{r|i}] @ G  (waves 0 and 4) ----
    v8f ax = splat8(bx);
    v8f xold;
    const bool xwave = (wq == 0);     // wave-uniform: EXEC stays all-ones
    if (xwave) {
#pragma unroll
      for (int kc = 0; kc < 22; ++kc) {
        const float* src; float scale;
        if (kc < 3)      { src = S      + mrow * S_STRIDE + kc * 4;            scale = p0;   }
        else if (kc < 6) { src = S + SG + mrow * S_STRIDE + (kc - 3) * 4;      scale = p0;   }
        else             { src = Sg     + mrow * S_STRIDE + 16 + (kc - 6) * 4; scale = 1.0f; }
        v2f a = *(const v2f*)(src + sel);
        a.x *= scale; a.y *= scale;
        const float* gk = fragG + (g * 22 + kc) * 64 + (lane << 1);
        ax = __builtin_amdgcn_wmma_f32_16x16x4_f32(false, a, false, *(const v2f*)gk, (short)0, ax, false, false);
      }
      const float* xsrc = Sg + (lane & 15);
#pragma unroll
      for (int v = 0; v < 8; ++v) {
        int mm = v + ((lane >> 4) << 3);
        xold[v] = xsrc[mm * S_STRIDE];
      }
    }
    __syncthreads();  // (C1) wave0/4 cross-reads of x done before writes
    if (xwave) {
      float* xdst = Sg + (lane & 15);
#pragma unroll
      for (int v = 0; v < 8; ++v) {
        int mm = v + ((lane >> 4) << 3);
        xdst[mm * S_STRIDE] = xold[v] + DTs * ax[v];   // cols>=10 stay 0
      }
    }
    __syncthreads();  // (C2) new x visible for next step
  }

  // ---- output: out[row, 0:10] = xr, out[row, 10:20] = xi ----
  for (int idx = tid; idx < 16 * 20; idx += 256) {
    int m = idx / 20, d = idx - m * 20;
    float v = (d < DD) ? S[m * S_STRIDE + d] : S[SG + m * S_STRIDE + (d - DD)];
    out[(blockIdx.x * 16 + m) * 20 + d] = v;
  }
}

extern "C" void kernel_launch(void* const* d_in, const int* in_sizes, int n_in,
                              void* d_out, int out_size, void* d_ws, size_t ws_size,
                              hipStream_t stream) {
  (void)in_sizes; (void)n_in; (void)out_size; (void)d_ws; (void)ws_size;
  const float* inputs = (const float*)d_in[0];
  const float* rho0_r = (const float*)d_in[1];
  const float* rho0_i = (const float*)d_in[2];
  const float* Ar     = (const float*)d_in[3];
  const float* Ai     = (const float*)d_in[4];
  const float* Wr     = (const float*)d_in[5];
  const float* Ur     = (const float*)d_in[6];
  const float* br     = (const float*)d_in[7];
  const float* Wi     = (const float*)d_in[8];
  const float* Ui     = (const float*)d_in[9];
  const float* bi     = (const float*)d_in[10];
  const float* Dar_w  = (const float*)d_in[11];
  const float* Dar_b  = (const float*)d_in[12];
  const float* Dai_w  = (const float*)d_in[13];
  const float* Dai_b  = (const float*)d_in[14];
  float* out = (float*)d_out;

  dim3 grid(BB / 16), block(256);
  size_t shmem = (size_t)SMEM_FLOATS * sizeof(float);   // ~186KB of 320KB/WGP
  euler_flex_rnn<<<grid, block, shmem, stream>>>(
      inputs, rho0_r, rho0_i, Ar, Ai, Wr, Ur, br, Wi, Ui, bi,
      Dar_w, Dar_b, Dai_w, Dai_b, out);
}